// CLASSNetLoss_66829691126278
// MI455X (gfx1250) — compile-verified
//
#include <hip/hip_runtime.h>
#include <hip/hip_bf16.h>

typedef __attribute__((ext_vector_type(2))) float v2f;
typedef __attribute__((ext_vector_type(8))) float v8f;

#define SS_CONST 210.0f  // sum(1..20)

// Wave-wide sum of one float per lane using V_WMMA_F32_16X16X4_F32.
// A (16x4): lane L<16 -> A[L][0]=x (VGPR0), A[L][1]=0 (VGPR1);
//           lane L>=16 -> A[L-16][2]=x, A[L-16][3]=0.
// B = ones(4x16)  =>  D[m][n] = x_m + x_{m+16} (row sum, all columns equal).
// D layout: VGPR r, lanes 0-15 hold M=r; lanes 16-31 hold M=8+r.
// Sum of 8 D VGPRs gives half-wave sums; one shfl_xor(16) completes it.
// Requires EXEC == all 1s: call only from fully convergent code.
__device__ __forceinline__ float wave_sum_wmma(float x) {
  v2f a; a.x = x;    a.y = 0.0f;
  v2f b; b.x = 1.0f; b.y = 1.0f;
  v8f c = {};
  c = __builtin_amdgcn_wmma_f32_16x16x4_f32(
      /*neg_a=*/false, a, /*neg_b=*/false, b,
      /*c_mod=*/(short)0, c, /*reuse_a=*/false, /*reuse_b=*/false);
  float s = ((c[0] + c[1]) + (c[2] + c[3])) + ((c[4] + c[5]) + (c[6] + c[7]));
  s += __shfl_xor(s, 16, 32);
  return s;  // uniform across the wave
}

__device__ __forceinline__ float focal_elem(float x, int t, float w) {
  float p   = fminf(fmaxf(x, 1e-5f), 0.99999f);
  float omp = 1.0f - p;
  float pos = -(SS_CONST - w) * logf(p)   * (omp * omp);
  float neg = -w              * logf(omp) * (p * p);
  return (t > 0) ? pos : neg;
}

__global__ void focal_partial_kernel(const float4* __restrict__ x4,
                                     const int4*   __restrict__ t4,
                                     const float*  __restrict__ xs,
                                     const int*    __restrict__ ts,
                                     float*        __restrict__ partials,
                                     int nvec, int tail) {
  const int tid    = blockIdx.x * blockDim.x + threadIdx.x;
  const int stride = gridDim.x * blockDim.x;

  float acc = 0.0f;
  for (int v = tid; v < nvec; v += stride) {
    float4 x = x4[v];
    int4   t = t4[v];
    if (v + stride < nvec) {
      __builtin_prefetch(&x4[v + stride], 0, 0);  // global_prefetch_b8
      __builtin_prefetch(&t4[v + stride], 0, 0);
    }
    // flat index = 4v; column = 4v % 20 = 4*(v%5); never wraps within a vec4
    float wb = (float)((v % 5) * 4 + 1);
    acc += focal_elem(x.x, t.x, wb);
    acc += focal_elem(x.y, t.y, wb + 1.0f);
    acc += focal_elem(x.z, t.z, wb + 2.0f);
    acc += focal_elem(x.w, t.w, wb + 3.0f);
  }

  // Scalar tail (N % 4), handled by one thread; reconverges before WMMA.
  if (tid == 0) {
    for (int j = 0; j < tail; ++j) {
      long long idx = (long long)nvec * 4 + j;
      int c = (int)(idx % 20);
      acc += focal_elem(xs[idx], ts[idx], (float)(c + 1));
    }
  }

  // Fully convergent wave reduction via WMMA
  float ws = wave_sum_wmma(acc);

  __shared__ float lds[32];
  const int lane = threadIdx.x & 31;
  const int wv   = threadIdx.x >> 5;
  if (lane == 0) lds[wv] = ws;
  __syncthreads();

  if (wv == 0) {
    const int nw = blockDim.x >> 5;
    float s = (lane < nw) ? lds[lane] : 0.0f;
    s += __shfl_xor(s, 1, 32);
    s += __shfl_xor(s, 2, 32);
    s += __shfl_xor(s, 4, 32);
    s += __shfl_xor(s, 8, 32);
    s += __shfl_xor(s, 16, 32);
    if (lane == 0) partials[blockIdx.x] = s;
  }
}

__global__ void focal_final_kernel(const float* __restrict__ partials, int n,
                                   float* __restrict__ out, float scale) {
  float acc = 0.0f;
  for (int i = threadIdx.x; i < n; i += blockDim.x) acc += partials[i];

  float ws = wave_sum_wmma(acc);

  __shared__ float lds[32];
  const int lane = threadIdx.x & 31;
  const int wv   = threadIdx.x >> 5;
  if (lane == 0) lds[wv] = ws;
  __syncthreads();

  if (threadIdx.x == 0) {
    const int nw = blockDim.x >> 5;
    float s = 0.0f;
    for (int w = 0; w < nw; ++w) s += lds[w];
    out[0] = s * scale;
  }
}

extern "C" void kernel_launch(void* const* d_in, const int* in_sizes, int n_in,
                              void* d_out, int out_size, void* d_ws, size_t ws_size,
                              hipStream_t stream) {
  const float* xs = (const float*)d_in[0];  // output: f32 [B, C]
  const int*   ts = (const int*)d_in[1];    // target: i32 [B, C]
  float* res      = (float*)d_out;          // scalar f32
  float* partials = (float*)d_ws;

  const long long N = (long long)in_sizes[0];  // B*C = 10,485,760
  const int nvec = (int)(N / 4);
  const int tail = (int)(N % 4);

  const int threads = 256;                  // 8 wave32 per block
  int blocks = (nvec + threads - 1) / threads;
  if (blocks > 2048) blocks = 2048;         // ~5 vec4 per thread at full size
  const int maxb = (int)(ws_size / sizeof(float));
  if (blocks > maxb) blocks = maxb;
  if (blocks < 1) blocks = 1;

  focal_partial_kernel<<<blocks, threads, 0, stream>>>(
      (const float4*)xs, (const int4*)ts, xs, ts, partials, nvec, tail);

  const float scale = 10.0f / (SS_CONST * (float)N);
  focal_final_kernel<<<1, 256, 0, stream>>>(partials, blocks, res, scale);
}